// CosineSimilarityGraphAttention_47382079209606
// MI455X (gfx1250) — compile-verified
//
#include <hip/hip_runtime.h>
#include <hip/hip_bf16.h>

#define NNODES   40000
#define NEDGES   640000
#define D        128
#define EPS_NORM 1e-8f
#define EPS_ATT  1e-8f

typedef __attribute__((ext_vector_type(2))) float v2f;
typedef __attribute__((ext_vector_type(8))) float v8f;

// ---- order-preserving float<->uint encoding for atomic segment-max ----
__device__ __forceinline__ unsigned encode_max(float f) {
    unsigned b = __float_as_uint(f);
    return (b & 0x80000000u) ? ~b : (b | 0x80000000u);
}
__device__ __forceinline__ float decode_max(unsigned k) {
    unsigned b = (k & 0x80000000u) ? (k ^ 0x80000000u) : ~k;
    return __uint_as_float(b);
}

// ---------------------------------------------------------------------
// Kernel 0: per-call initialization (out = 0, att_sum = 0, segmax = enc(-inf))
// ---------------------------------------------------------------------
__global__ void cga_init(float* __restrict__ out,
                         unsigned* __restrict__ segmax,
                         float* __restrict__ attsum) {
    int i = blockIdx.x * blockDim.x + threadIdx.x;
    if (i < NNODES * D) out[i] = 0.0f;
    if (i < NNODES) {
        segmax[i] = 0x007FFFFFu;   // encode_max(-inf)
        attsum[i] = 0.0f;
    }
}

// ---------------------------------------------------------------------
// Kernel 1: H = X(40000x128) @ W(128x128) via V_WMMA_F32_16X16X4_F32.
// One block = 16 rows of X; 8 waves each own one 16-wide N tile.
// W staged in 64KB LDS (WGP has 320KB). K loop: 32 WMMA ops of K=4.
// A frag (16x4 f32): lanes 0-15 -> K=0,1 ; lanes 16-31 -> K=2,3 (per lane: a.x,a.y)
// B frag (4x16 f32): same K split, N = lane%16
// C/D  (16x16 f32): VGPR j: lanes 0-15 -> M=j, lanes 16-31 -> M=j+8
// ---------------------------------------------------------------------
__global__ void __launch_bounds__(256)
cga_gemm_wmma(const float* __restrict__ X, const float* __restrict__ W,
              float* __restrict__ H) {
    __shared__ float lW[D * D];          // 64 KB
    for (int i = threadIdx.x; i < (D * D) / 4; i += 256)
        ((float4*)lW)[i] = ((const float4*)W)[i];
    __syncthreads();

    const int lane = threadIdx.x & 31;
    const int wave = threadIdx.x >> 5;
    const int m0   = blockIdx.x * 16;
    const int n0   = wave * 16;
    const int r    = lane & 15;
    const int kh   = (lane >> 4) << 1;   // 0 or 2

    const float* xrow = X + (m0 + r) * D;
    v8f acc = {};
    #pragma unroll
    for (int k = 0; k < D; k += 4) {
        v2f a, b;
        a.x = xrow[k + kh];
        a.y = xrow[k + kh + 1];
        b.x = lW[(k + kh) * D + n0 + r];
        b.y = lW[(k + kh + 1) * D + n0 + r];
        acc = __builtin_amdgcn_wmma_f32_16x16x4_f32(
            /*neg_a=*/false, a, /*neg_b=*/false, b,
            /*c_mod=*/(short)0, acc, /*reuse_a=*/false, /*reuse_b=*/false);
    }

    const int mo = (lane >> 4) << 3;     // 0 or 8
    #pragma unroll
    for (int j = 0; j < 8; ++j)
        H[(m0 + mo + j) * D + n0 + r] = acc[j];
}

// ---------------------------------------------------------------------
// Kernel 2: per-node 1/sqrt(max(||h||^2, eps^2)). One wave32 per node,
// float4 per lane (32*4 = 128), shfl-xor tree reduction.
// ---------------------------------------------------------------------
__global__ void cga_invnorm(const float* __restrict__ H, float* __restrict__ invn) {
    const int node = (blockIdx.x * blockDim.x + threadIdx.x) >> 5;
    const int lane = threadIdx.x & 31;
    if (node >= NNODES) return;
    float4 v = ((const float4*)(H + (long)node * D))[lane];
    float s = v.x * v.x + v.y * v.y + v.z * v.z + v.w * v.w;
    #pragma unroll
    for (int off = 16; off; off >>= 1) s += __shfl_xor(s, off, 32);
    if (lane == 0) invn[node] = 1.0f / sqrtf(fmaxf(s, EPS_NORM * EPS_NORM));
}

// ---------------------------------------------------------------------
// Kernel 3: per-edge cosine + atomic segment-max. One wave32 per edge.
// h rows are L2-resident (20.5MB << 192MB L2).
// ---------------------------------------------------------------------
__global__ void cga_edge_cos(const float* __restrict__ H,
                             const long long* __restrict__ edges,
                             const float* __restrict__ invn,
                             float* __restrict__ cosv,
                             unsigned* __restrict__ segmax) {
    const int e    = (blockIdx.x * blockDim.x + threadIdx.x) >> 5;
    const int lane = threadIdx.x & 31;
    if (e >= NEDGES) return;
    const int t = (int)edges[2 * e];
    const int s = (int)edges[2 * e + 1];
    float4 a = ((const float4*)(H + (long)t * D))[lane];
    float4 b = ((const float4*)(H + (long)s * D))[lane];
    float d = a.x * b.x + a.y * b.y + a.z * b.z + a.w * b.w;
    #pragma unroll
    for (int off = 16; off; off >>= 1) d += __shfl_xor(d, off, 32);
    if (lane == 0) {
        float c = d * invn[t] * invn[s];
        cosv[e] = c;
        atomicMax(&segmax[t], encode_max(c));
    }
}

// ---------------------------------------------------------------------
// Kernel 4: exp(cos - segmax[t]) and segment-sum via fp32 L2 atomics.
// ---------------------------------------------------------------------
__global__ void cga_edge_exp(const long long* __restrict__ edges,
                             const float* __restrict__ cosv,
                             const unsigned* __restrict__ segmax,
                             float* __restrict__ exps,
                             float* __restrict__ attsum) {
    int e = blockIdx.x * blockDim.x + threadIdx.x;
    if (e >= NEDGES) return;
    int t = (int)edges[2 * e];
    float m = decode_max(segmax[t]);
    float v = expf(cosv[e] - m);
    exps[e] = v;
    atomicAdd(&attsum[t], v);
}

// ---------------------------------------------------------------------
// Kernel 5: out[t] += attn * h[s]. One wave32 per edge; 4 fp32 atomics
// per lane -> GLOBAL_ATOMIC_ADD_F32 serviced in L2 (out is 20.5MB, resident).
// ---------------------------------------------------------------------
__global__ void cga_edge_scatter(const float* __restrict__ H,
                                 const long long* __restrict__ edges,
                                 const float* __restrict__ exps,
                                 const float* __restrict__ attsum,
                                 float* __restrict__ out) {
    const int e    = (blockIdx.x * blockDim.x + threadIdx.x) >> 5;
    const int lane = threadIdx.x & 31;
    if (e >= NEDGES) return;
    const int t = (int)edges[2 * e];
    const int s = (int)edges[2 * e + 1];
    const float attn = exps[e] / (attsum[t] + EPS_ATT);
    float4 v = ((const float4*)(H + (long)s * D))[lane];
    float* o = out + (long)t * D + lane * 4;
    atomicAdd(o + 0, v.x * attn);
    atomicAdd(o + 1, v.y * attn);
    atomicAdd(o + 2, v.z * attn);
    atomicAdd(o + 3, v.w * attn);
}

extern "C" void kernel_launch(void* const* d_in, const int* in_sizes, int n_in,
                              void* d_out, int out_size, void* d_ws, size_t ws_size,
                              hipStream_t stream) {
    const float*     X     = (const float*)d_in[0];       // node_states 40000x128 f32
    const long long* edges = (const long long*)d_in[1];   // edges 640000x2 i64
    const float*     W     = (const float*)d_in[2];       // kernel 128x128 f32
    float* out = (float*)d_out;                           // 40000x128 f32

    // workspace layout (floats): H | invn | segmax | attsum | cosv | exps  ~26MB
    float*    H      = (float*)d_ws;
    float*    invn   = H + (size_t)NNODES * D;
    unsigned* segmax = (unsigned*)(invn + NNODES);
    float*    attsum = (float*)(segmax + NNODES);
    float*    cosv   = attsum + NNODES;
    float*    exps   = cosv + NEDGES;

    cga_init<<<(NNODES * D + 255) / 256, 256, 0, stream>>>(out, segmax, attsum);
    cga_gemm_wmma<<<NNODES / 16, 256, 0, stream>>>(X, W, H);
    cga_invnorm<<<(NNODES * 32 + 255) / 256, 256, 0, stream>>>(H, invn);
    cga_edge_cos<<<(NEDGES * 32 + 255) / 256, 256, 0, stream>>>(H, edges, invn, cosv, segmax);
    cga_edge_exp<<<(NEDGES + 255) / 256, 256, 0, stream>>>(edges, cosv, segmax, exps, attsum);
    cga_edge_scatter<<<(NEDGES * 32 + 255) / 256, 256, 0, stream>>>(H, edges, exps, attsum, out);
}